// MyLinear_9036611191230
// MI455X (gfx1250) — compile-verified
//
#include <hip/hip_runtime.h>

typedef __attribute__((ext_vector_type(2))) float v2f;
typedef __attribute__((ext_vector_type(8))) float v8f;

#define BB   16
#define NN   512
#define TT   128
#define DIN  64
#define DOUT 64
#define PAD  68   // padded LDS row stride (floats) -> conflict-free b64 frag reads

__global__ __launch_bounds__(256) void node_linear_wmma_f32(
    const float* __restrict__ x,     // [B,N,T,DIN]
    const float* __restrict__ W,     // [N,DIN,DOUT]
    const float* __restrict__ bias,  // [N,DOUT]
    float* __restrict__ out)         // [B,N,T,DOUT]
{
    __shared__ float ldsX[TT * PAD];     // x tile, row-major [t][d], padded
    __shared__ float ldsWt[DOUT * PAD];  // W transposed [e][d], padded
    __shared__ float ldsB[DOUT];

    const int bn  = blockIdx.x;          // 0..B*N-1 ; (b,n) slab is contiguous in x/out
    const int n   = bn & (NN - 1);
    const int tid = threadIdx.x;

    const float* xg = x + (size_t)bn * (TT * DIN);       // 8192 floats, contiguous
    const float* Wg = W + (size_t)n  * (DIN * DOUT);     // 4096 floats
    float*       og = out + (size_t)bn * (TT * DOUT);

    // ---- stage x tile: 8192 floats via b128 loads (coalesced HBM stream) ----
    {
        const float4* xg4 = reinterpret_cast<const float4*>(xg);
#pragma unroll
        for (int i = 0; i < 8; ++i) {
            int idx4 = tid + i * 256;            // 0..2047
            float4 v = xg4[idx4];
            int e   = idx4 << 2;
            int row = e >> 6;                    // t within tile
            int col = e & 63;                    // d
            float* p = &ldsX[row * PAD + col];
            p[0] = v.x; p[1] = v.y; p[2] = v.z; p[3] = v.w;
        }
    }
    // ---- stage W transposed: ldsWt[e][d] = W[d][e]  (W is L2-resident) ----
    {
#pragma unroll
        for (int i = 0; i < 16; ++i) {
            int idx = tid + i * 256;             // 0..4095
            int d   = idx >> 6;
            int e   = idx & 63;
            ldsWt[e * PAD + d] = Wg[idx];
        }
    }
    if (tid < DOUT) ldsB[tid] = bias[n * DOUT + tid];

    __syncthreads();

    const int wave = tid >> 5;     // 0..7 -> 16-row slab
    const int lane = tid & 31;
    const int lo   = lane & 15;    // row (A) / column (B,C)
    const int hi   = lane >> 4;    // selects K pair (+2) and M (+8) halves
    const int m0   = wave << 4;

    // C init = bias broadcast over rows (C/D layout: VGPR v -> M = v + 8*hi, N = lo)
    v8f acc[4];
#pragma unroll
    for (int nt = 0; nt < 4; ++nt) {
        float bv = ldsB[nt * 16 + lo];
#pragma unroll
        for (int v = 0; v < 8; ++v) acc[nt][v] = bv;
    }

    // A frag (16x4 f32): lane(lo,hi) -> {x[m0+lo][4k+2hi], x[m0+lo][4k+2hi+1]}
    // B frag (4x16 f32): lane(lo,hi) -> {W[4k+2hi][nt*16+lo], W[4k+2hi+1][nt*16+lo]}
    const float* aBase = &ldsX [(m0 + lo) * PAD + 2 * hi];
    const float* bBase = &ldsWt[ lo       * PAD + 2 * hi];

#pragma unroll
    for (int ks = 0; ks < 16; ++ks) {
        v2f a = *reinterpret_cast<const v2f*>(aBase + 4 * ks);   // ds_load_b64
#pragma unroll
        for (int nt = 0; nt < 4; ++nt) {
            v2f b = *reinterpret_cast<const v2f*>(bBase + nt * (16 * PAD) + 4 * ks);
            acc[nt] = __builtin_amdgcn_wmma_f32_16x16x4_f32(
                /*neg_a=*/false, a, /*neg_b=*/false, b,
                /*c_mod=*/(short)0, acc[nt], /*reuse_a=*/false, /*reuse_b=*/false);
        }
    }

    // ---- store: per half-wave, 16 consecutive floats -> coalesced 64B segments ----
#pragma unroll
    for (int nt = 0; nt < 4; ++nt) {
#pragma unroll
        for (int v = 0; v < 8; ++v) {
            int m = m0 + v + 8 * hi;
            og[m * DOUT + nt * 16 + lo] = acc[nt][v];
        }
    }
}

extern "C" void kernel_launch(void* const* d_in, const int* in_sizes, int n_in,
                              void* d_out, int out_size, void* d_ws, size_t ws_size,
                              hipStream_t stream) {
    const float* x    = (const float*)d_in[0];   // [16,512,128,64]
    const float* W    = (const float*)d_in[1];   // [512,64,64]
    const float* bias = (const float*)d_in[2];   // [512,64]
    float*       out  = (float*)d_out;           // [16,512,128,64]

    dim3 grid(BB * NN);   // 8192 blocks, one per (b,n)
    dim3 block(256);      // 8 wave32s
    node_linear_wmma_f32<<<grid, block, 0, stream>>>(x, W, bias, out);
}